// GCNConv_79302276153382
// MI455X (gfx1250) — compile-verified
//
#include <hip/hip_runtime.h>
#include <hip/hip_bf16.h>

#define D 128   // D_IN == D_OUT == 128 in the reference

typedef __attribute__((ext_vector_type(2))) float v2f;
typedef __attribute__((ext_vector_type(8))) float v8f;

// ---------------- degree = 1 (self loop) + in-degree -------------------
__global__ void deg_init_kernel(float* __restrict__ deg, int N) {
  int i = blockIdx.x * blockDim.x + threadIdx.x;
  if (i < N) deg[i] = 1.0f;
}

__global__ void deg_count_kernel(const long long* __restrict__ dst,
                                 float* __restrict__ deg, int E) {
  int e = blockIdx.x * blockDim.x + threadIdx.x;
  if (e < E) atomicAdd(&deg[(int)dst[e]], 1.0f);
}

__global__ void rsqrt_kernel(float* __restrict__ deg, int N) {
  int i = blockIdx.x * blockDim.x + threadIdx.x;
  if (i < N) deg[i] = rsqrtf(deg[i]);   // deg >= 1 always (self-loops)
}

// ---------------- h = x @ W via V_WMMA_F32_16X16X4_F32 -----------------
// One wave computes one 16x16 tile of H. K loop: 128/4 = 32 WMMAs.
// A 16x4 layout (ISA 7.12.2): lanes 0-15 -> row M=lane, K = k+{0,1};
//                             lanes16-31 -> row M=lane-16, K = k+{2,3}.
// B 4x16: lanes 0-15 -> col N=lane, K = k+{0,1}; lanes16-31 -> col N=lane-16, K=k+{2,3}.
// C/D 16x16: VGPR r holds M = r + 8*(lane>=16), N = lane&15.
__global__ void gemm_wmma_kernel(const float* __restrict__ X,
                                 const float* __restrict__ W,
                                 float* __restrict__ H, int mtiles) {
  const int wave  = (int)((blockIdx.x * blockDim.x + threadIdx.x) >> 5);
  const int lane  = threadIdx.x & 31;
  const int mtile = wave >> 3;        // D/16 == 8 N-tiles
  const int ntile = wave & 7;
  if (mtile >= mtiles) return;        // wave-uniform guard: EXEC stays all-1s

  const int lhalf = lane >> 4;        // 0 or 1
  const int l16   = lane & 15;
  const int row0  = mtile * 16;
  const int col0  = ntile * 16;

  v8f acc = {0.f, 0.f, 0.f, 0.f, 0.f, 0.f, 0.f, 0.f};

  for (int k = 0; k < D; k += 4) {
    const float* ap = X + (size_t)(row0 + l16) * D + (k + 2 * lhalf);
    v2f a; a.x = ap[0]; a.y = ap[1];                        // global_load_b64

    const float* bp = W + (size_t)(k + 2 * lhalf) * D + (col0 + l16);
    v2f b; b.x = bp[0]; b.y = bp[D];                        // two strided b32

    acc = __builtin_amdgcn_wmma_f32_16x16x4_f32(
        /*neg_a=*/false, a, /*neg_b=*/false, b,
        /*c_mod=*/(short)0, acc, /*reuse_a=*/false, /*reuse_b=*/false);
  }

  float* outp = H + (size_t)row0 * D + col0;
#pragma unroll
  for (int r = 0; r < 8; ++r)
    outp[(size_t)(r + 8 * lhalf) * D + l16] = acc[r];
}

// ---------------- out = bias + dis[n]^2 * h[n]  (self-loop term) -------
__global__ void out_init_kernel(const float* __restrict__ H,
                                const float* __restrict__ dis,
                                const float* __restrict__ bias,
                                float* __restrict__ out, int N) {
  int idx = blockIdx.x * blockDim.x + threadIdx.x;
  int n = idx >> 7;      // / 128
  int d = idx & 127;
  if (n < N) {
    float s = dis[n];
    out[idx] = bias[d] + s * s * H[idx];
  }
}

// ---------------- out[dst] += dis[src]*dis[dst] * h[src] ---------------
// One wave32 per edge; float4 coalesced gather (512B row), 4 f32 atomics/lane.
__global__ void edge_scatter_kernel(const long long* __restrict__ src,
                                    const long long* __restrict__ dst,
                                    const float* __restrict__ dis,
                                    const float* __restrict__ H,
                                    float* __restrict__ out, int E) {
  int wid  = (int)((blockIdx.x * blockDim.x + threadIdx.x) >> 5);
  int lane = threadIdx.x & 31;
  if (wid >= E) return;               // wave-uniform

  int s = (int)src[wid];
  int d = (int)dst[wid];
  float norm = dis[s] * dis[d];

  const float4* hs = (const float4*)(H + (size_t)s * D);
  float4 v = hs[lane];                // global_load_b128, coalesced
  float* od = out + (size_t)d * D + lane * 4;
  atomicAdd(od + 0, norm * v.x);
  atomicAdd(od + 1, norm * v.y);
  atomicAdd(od + 2, norm * v.z);
  atomicAdd(od + 3, norm * v.w);
}

extern "C" void kernel_launch(void* const* d_in, const int* in_sizes, int n_in,
                              void* d_out, int out_size, void* d_ws, size_t ws_size,
                              hipStream_t stream) {
  const float*     x  = (const float*)d_in[0];              // [N, 128]
  const long long* ei = (const long long*)d_in[1];          // [2, E] int64
  const float*     W  = (const float*)d_in[2];              // [128, 128]
  const float*     b  = (const float*)d_in[3];              // [128]
  float*           out = (float*)d_out;                     // [N, 128]

  const int N = in_sizes[0] / D;
  const int E = in_sizes[1] / 2;
  const long long* src = ei;
  const long long* dst = ei + E;

  // workspace carve: deg/dis (N floats, reused in place) then H (N*D floats)
  char* ws = (char*)d_ws;
  float* deg = (float*)ws;                                  // becomes dis in place
  size_t deg_bytes = ((size_t)N * sizeof(float) + 255) & ~(size_t)255;
  float* H = (float*)(ws + deg_bytes);

  const int T = 256;

  // 1) deg = 1.0 (self loop)
  deg_init_kernel<<<(N + T - 1) / T, T, 0, stream>>>(deg, N);
  // 2) deg[dst] += 1 per edge
  deg_count_kernel<<<(E + T - 1) / T, T, 0, stream>>>(dst, deg, E);
  // 3) dis = rsqrt(deg), in place
  rsqrt_kernel<<<(N + T - 1) / T, T, 0, stream>>>(deg, N);

  // 4) H = x @ W  (fp32 WMMA); one wave per 16x16 tile
  const int mtiles = (N + 15) / 16;                         // 3125
  const long long waves = (long long)mtiles * (D / 16);     // 25000
  const int gemm_blocks = (int)((waves * 32 + T - 1) / T);  // 3125
  gemm_wmma_kernel<<<gemm_blocks, T, 0, stream>>>(x, W, H, mtiles);

  // 5) out = bias + dis^2 * H   (initializes every element of d_out)
  const long long total = (long long)N * D;
  out_init_kernel<<<(int)((total + T - 1) / T), T, 0, stream>>>(H, deg, b, out, N);

  // 6) scatter-add over edges
  const int edge_blocks = (int)(((long long)E * 32 + T - 1) / T);
  edge_scatter_kernel<<<edge_blocks, T, 0, stream>>>(src, dst, deg, H, out, E);
}